// PredictionInterAttention_90744069030182
// MI455X (gfx1250) — compile-verified
//
#include <hip/hip_runtime.h>
#include <hip/hip_bf16.h>

typedef __attribute__((ext_vector_type(2))) float v2f;
typedef __attribute__((ext_vector_type(8))) float v8f;

#define B_ 8
#define C_ 32
#define T_ 9
#define L_ 512
#define H_ 8
#define D_ 4

// ---------------------------------------------------------------------------
// Kernel 1: q/k/v projections + L2 norm over head_dim.
// One thread per (b,t,l,h): 3 proj x 4 outputs x 32 MACs. Grid = 294912/256.
// q/k/v layout: (B,T,H,L,D) contiguous.
// ---------------------------------------------------------------------------
__global__ __launch_bounds__(256) void qkv_kernel(
    const float* __restrict__ x,
    const float* __restrict__ Wq, const float* __restrict__ bq,
    const float* __restrict__ Wk, const float* __restrict__ bk,
    const float* __restrict__ Wv, const float* __restrict__ bv,
    float* __restrict__ q, float* __restrict__ k, float* __restrict__ v)
{
    int tid = blockIdx.x * 256 + threadIdx.x;        // 0..294911
    int h   = tid & 7;
    int l   = (tid >> 3) & 511;
    int bt  = tid >> 12;                             // 0..71  (= b*T + t)
    int b   = bt / T_;
    int t   = bt - b * T_;

    // gather xi[c] = x[b,c,t,l]  (stride T*L between channels)
    float xv[C_];
    int base = ((b * C_) * T_ + t) * L_ + l;
    #pragma unroll
    for (int c = 0; c < C_; ++c) xv[c] = x[base + c * (T_ * L_)];

    const float* Ws[3] = { Wq, Wk, Wv };
    const float* bs[3] = { bq, bk, bv };
    float* outs[3]     = { q, k, v };

    size_t obase = ((size_t)(bt * H_ + h) * L_ + l) * D_;
    for (int p = 0; p < 3; ++p) {
        float yv[D_];
        float ss = 0.f;
        #pragma unroll
        for (int d = 0; d < D_; ++d) {
            int row = h * D_ + d;
            float acc = bs[p][row];
            const float* wrow = Ws[p] + row * C_;
            #pragma unroll
            for (int c = 0; c < C_; ++c) acc += xv[c] * wrow[c];
            yv[d] = acc;
            ss += acc * acc;
        }
        float nrm = fmaxf(sqrtf(ss), 1e-12f);
        float inv = 1.0f / nrm;
        #pragma unroll
        for (int d = 0; d < D_; ++d) outs[p][obase + d] = yv[d] * inv;
    }
}

// ---------------------------------------------------------------------------
// Kernel 2: fused attention.
// grid = B*T*(L/16) = 2304 blocks, 256 threads = 8 waves.
// Wave w: (a) computes raw logits for head h=w via v_wmma_f32_16x16x4_f32,
//         (b) owns mixed-head H=w for softmax + value accumulation.
// Two-pass softmax (max pass, then exp/accumulate pass; raw logits are
// recomputed by WMMA in pass 2 — compute is free at this size).
// Writes x_att (B,T,L,C) with c = h2*4 + d.
// ---------------------------------------------------------------------------
__global__ __launch_bounds__(256) void attn_kernel(
    const float* __restrict__ q, const float* __restrict__ k,
    const float* __restrict__ v,
    const float* __restrict__ Wl,   // W_logit (H,H)
    const float* __restrict__ Wc,   // W_ctx   (H,H)
    const float* __restrict__ rpb,  // rel_pos_bias (H, 2L-1=1023)
    float* __restrict__ x_att)
{
    __shared__ float raw[H_][16][64];    // 32 KB: raw logits chunk, all heads
    __shared__ float vsh[H_][64][D_];    //  8 KB: v chunk, all heads
    __shared__ float xpart[H_][16][32];  // 16 KB: per-wave output partials

    const int blk   = blockIdx.x;
    const int itile = blk & 31;
    const int bt    = blk >> 5;          // b*T + t
    const int i0    = itile * 16;
    const int tid   = threadIdx.x;
    const int w     = tid >> 5;          // wave id == head index
    const int lane  = tid & 31;
    const int n     = lane & 15;
    const int hi    = lane >> 4;

    const size_t hdbase = (size_t)(bt * H_ + w) * L_ * D_;

    // A operand (q tile, 16x4 f32): lane m=n holds K=2*hi, 2*hi+1 of row i0+m.
    v2f aq;
    {
        const float* qp = q + hdbase + (size_t)(i0 + n) * D_ + hi * 2;
        aq.x = qp[0]; aq.y = qp[1];
    }
    // mixing weights
    float wl[H_], wc[H_];
    #pragma unroll
    for (int h = 0; h < H_; ++h)  wl[h] = Wl[h * H_ + w];   // W_logit[h][w]
    #pragma unroll
    for (int h2 = 0; h2 < H_; ++h2) wc[h2] = Wc[w * H_ + h2]; // W_ctx[w][h2]

    const int i_row = n;     // local row this lane reduces
    const int jhalf = hi;    // which 32-wide half of the 64 chunk

    // compute raw logits chunk (head w) into raw[w][.][.]
    auto compute_raw = [&](int ch) {
        #pragma unroll
        for (int jt = 0; jt < 4; ++jt) {
            const int j0 = ch * 64 + jt * 16;
            v2f bk;
            const float* kp = k + hdbase + (size_t)(j0 + n) * D_ + hi * 2;
            bk.x = kp[0]; bk.y = kp[1];
            v8f cc = {};
            cc = __builtin_amdgcn_wmma_f32_16x16x4_f32(
                false, aq, false, bk, (short)0, cc, false, false);
            #pragma unroll
            for (int r = 0; r < 8; ++r) {
                const int il = r + 8 * hi;
                const int ig = i0 + il;
                const int jg = j0 + n;
                raw[w][il][jt * 16 + n] =
                    cc[r] * 0.5f + rpb[w * 1023 + (ig - jg + 511)];
            }
        }
    };

    // ----------------- pass 1: row max of mixed logits -----------------
    float pmax = -3.0e38f;
    for (int ch = 0; ch < 8; ++ch) {
        __syncthreads();
        compute_raw(ch);
        __syncthreads();
        for (int jj = 0; jj < 32; ++jj) {
            const int jl = jhalf * 32 + jj;
            float m = 0.f;
            #pragma unroll
            for (int h = 0; h < H_; ++h) m += wl[h] * raw[h][i_row][jl];
            pmax = fmaxf(pmax, m);
        }
    }
    float rmax = fmaxf(pmax, __shfl_xor(pmax, 16));

    // ----------------- pass 2: exp-sum + value accumulation -------------
    float acc[32];
    #pragma unroll
    for (int z = 0; z < 32; ++z) acc[z] = 0.f;
    float psum = 0.f;

    for (int ch = 0; ch < 8; ++ch) {
        __syncthreads();
        compute_raw(ch);
        // cooperative load of v chunk for all 8 heads: 2048 floats
        #pragma unroll
        for (int r = 0; r < 8; ++r) {
            int e  = r * 256 + tid;
            int h2 = e >> 8;
            int jl = (e >> 2) & 63;
            int d  = e & 3;
            vsh[h2][jl][d] =
                v[((size_t)(bt * H_ + h2) * L_ + ch * 64 + jl) * D_ + d];
        }
        __syncthreads();
        for (int jj = 0; jj < 32; ++jj) {
            const int jl = jhalf * 32 + jj;
            float m = 0.f;
            #pragma unroll
            for (int h = 0; h < H_; ++h) m += wl[h] * raw[h][i_row][jl];
            float p = __expf(m - rmax);
            psum += p;
            #pragma unroll
            for (int h2 = 0; h2 < H_; ++h2)
                #pragma unroll
                for (int d = 0; d < D_; ++d)
                    acc[h2 * 4 + d] += p * vsh[h2][jl][d];
        }
    }
    float tot = psum + __shfl_xor(psum, 16);
    float inv = 1.0f / tot;
    #pragma unroll
    for (int z = 0; z < 32; ++z) acc[z] += __shfl_xor(acc[z], 16);

    if (lane < 16) {
        #pragma unroll
        for (int z = 0; z < 32; ++z)
            xpart[w][i_row][z] = wc[z >> 2] * acc[z] * inv;
    }
    __syncthreads();

    // reduce over waves (mixed head index) and write x_att
    #pragma unroll
    for (int r = 0; r < 2; ++r) {
        int e = r * 256 + tid;         // 0..511  -> (i local, c)
        int i = e >> 5;
        int z = e & 31;
        float s = 0.f;
        #pragma unroll
        for (int h = 0; h < H_; ++h) s += xpart[h][i][z];
        x_att[((size_t)bt * L_ + i0 + i) * C_ + z] = s;
    }
}

// ---------------------------------------------------------------------------
// Kernel 3: xm = (x_att @ Wm^T + bm) written in (B,C,T,L) layout.
// WMMA f32 16x16x4, 8 K-steps. One 16x16 tile per wave; grid = 576 blocks.
// ---------------------------------------------------------------------------
__global__ __launch_bounds__(256) void mm_wm_kernel(
    const float* __restrict__ xa, const float* __restrict__ Wm,
    const float* __restrict__ bm, float* __restrict__ xm)
{
    const int tid  = threadIdx.x;
    const int w    = tid >> 5, lane = tid & 31;
    const int n    = lane & 15, hi = lane >> 4;
    const int wt   = blockIdx.x * 8 + w;     // 0..4607
    const int mt   = wt >> 1;
    const int n0   = (wt & 1) * 16;
    const int row0 = mt * 16;

    v8f c = {};
    #pragma unroll
    for (int ks = 0; ks < 8; ++ks) {
        const int kk = ks * 4 + hi * 2;
        v2f a, b;
        const float* ap = xa + (size_t)(row0 + n) * C_ + kk;
        a.x = ap[0]; a.y = ap[1];
        const float* bp = Wm + (size_t)(n0 + n) * C_ + kk;
        b.x = bp[0]; b.y = bp[1];
        c = __builtin_amdgcn_wmma_f32_16x16x4_f32(
            false, a, false, b, (short)0, c, false, false);
    }
    const int cch = n0 + n;
    const float bias = bm[cch];
    #pragma unroll
    for (int r = 0; r < 8; ++r) {
        int m   = row0 + r + 8 * hi;         // global row = (b*T+t)*L + l
        int b_  = m / (T_ * L_);
        int rem = m - b_ * (T_ * L_);
        int t   = rem >> 9;
        int l   = rem & 511;
        xm[(((size_t)b_ * C_ + cch) * T_ + t) * L_ + l] = c[r] + bias;
    }
}

// ---------------------------------------------------------------------------
// Kernel 4: conv (K=C*T=288 MACs) + BN(eval) + ReLU -> y (B,C,L). Grid=512.
// ---------------------------------------------------------------------------
__global__ __launch_bounds__(256) void conv_kernel(
    const float* __restrict__ xm, const float* __restrict__ pp,
    const float* __restrict__ cw, const float* __restrict__ cb,
    const float* __restrict__ bg, const float* __restrict__ bb,
    float* __restrict__ y)
{
    int e  = blockIdx.x * 256 + threadIdx.x;   // 0..131071
    int l  = e & 511;
    int co = (e >> 9) & 31;
    int b  = e >> 14;
    float acc = cb[co];
    for (int ci = 0; ci < C_; ++ci) {
        const float* wrow = cw + (co * C_ + ci) * T_;
        const float* xrow = xm + (((size_t)b * C_ + ci) * T_) * L_ + l;
        #pragma unroll
        for (int t = 0; t < T_ - 1; ++t) acc += xrow[t * L_] * wrow[t];
        acc += pp[ci * L_ + l] * wrow[T_ - 1];
    }
    acc *= 0.99999500003750f;                  // 1/sqrt(1+1e-5)
    acc = acc * bg[co] + bb[co];
    y[e] = fmaxf(acc, 0.f);
}

// ---------------------------------------------------------------------------
// Kernel 5: p = y @ pl_w^T + pl_b, fused out[:,:,8,:] = xm[:,:,8,:] - p.
// WMMA f32 16x16x4 over K=512 (128 steps). M=256 rows (b*C+c), N=512 (l).
// grid = 64 blocks x 8 waves = 512 tiles.
// ---------------------------------------------------------------------------
__global__ __launch_bounds__(256) void pl_kernel(
    const float* __restrict__ y, const float* __restrict__ plw,
    const float* __restrict__ plb, const float* __restrict__ xm,
    float* __restrict__ out)
{
    const int tid  = threadIdx.x;
    const int w    = tid >> 5, lane = tid & 31;
    const int n    = lane & 15, hi = lane >> 4;
    const int wt   = blockIdx.x * 8 + w;     // 0..511
    const int mt   = wt >> 5;                // 0..15
    const int n0   = (wt & 31) * 16;
    const int row0 = mt * 16;

    v8f c = {};
    for (int ks = 0; ks < 128; ++ks) {
        const int kk = ks * 4 + hi * 2;
        v2f a, b;
        const float* ap = y + (size_t)(row0 + n) * L_ + kk;
        a.x = ap[0]; a.y = ap[1];
        const float* bp = plw + (size_t)(n0 + n) * L_ + kk;
        b.x = bp[0]; b.y = bp[1];
        c = __builtin_amdgcn_wmma_f32_16x16x4_f32(
            false, a, false, b, (short)0, c, false, false);
    }
    const int lcol = n0 + n;
    const float bias = plb[lcol];
    #pragma unroll
    for (int r = 0; r < 8; ++r) {
        int m  = row0 + r + 8 * hi;          // row = b*C + c
        int b_ = m >> 5, cc = m & 31;
        size_t oidx = (((size_t)b_ * C_ + cc) * T_ + (T_ - 1)) * L_ + lcol;
        out[oidx] = xm[oidx] - (c[r] + bias);
    }
}

// ---------------------------------------------------------------------------
// Kernel 6: passthrough copy of t<8 slices. Grid = 4096.
// ---------------------------------------------------------------------------
__global__ __launch_bounds__(256) void copy_kernel(
    const float* __restrict__ xm, float* __restrict__ out)
{
    int e  = blockIdx.x * 256 + threadIdx.x;   // 0..1048575
    int l  = e & 511;
    int r  = e >> 9;
    int t  = r & 7;  r >>= 3;
    int cc = r & 31;
    int b  = r >> 5;
    size_t idx = (((size_t)b * C_ + cc) * T_ + t) * L_ + l;
    out[idx] = xm[idx];
}

// ---------------------------------------------------------------------------
extern "C" void kernel_launch(void* const* d_in, const int* in_sizes, int n_in,
                              void* d_out, int out_size, void* d_ws, size_t ws_size,
                              hipStream_t stream)
{
    const float* x   = (const float*)d_in[0];
    const float* Wq  = (const float*)d_in[1];
    const float* bq  = (const float*)d_in[2];
    const float* Wk  = (const float*)d_in[3];
    const float* bk  = (const float*)d_in[4];
    const float* Wv  = (const float*)d_in[5];
    const float* bv  = (const float*)d_in[6];
    const float* Wm  = (const float*)d_in[7];
    const float* bm  = (const float*)d_in[8];
    const float* Wl  = (const float*)d_in[9];
    const float* Wc  = (const float*)d_in[10];
    const float* rpb = (const float*)d_in[11];
    const float* pp  = (const float*)d_in[12];
    const float* cw  = (const float*)d_in[13];
    const float* cb  = (const float*)d_in[14];
    const float* bg  = (const float*)d_in[15];
    const float* bb  = (const float*)d_in[16];
    const float* plw = (const float*)d_in[17];
    const float* plb = (const float*)d_in[18];
    float* out = (float*)d_out;

    const size_t NQ = (size_t)B_ * T_ * H_ * L_ * D_;   // 1179648
    float* ws = (float*)d_ws;
    float* q  = ws;
    float* k  = q  + NQ;
    float* v  = k  + NQ;
    float* xa = v  + NQ;        // (B*T*L, C)
    float* xm = xa + NQ;        // (B,C,T,L)
    float* yb = xm + NQ;        // (B,C,L) = 131072

    qkv_kernel <<<1152, 256, 0, stream>>>(x, Wq, bq, Wk, bk, Wv, bv, q, k, v);
    attn_kernel<<<2304, 256, 0, stream>>>(q, k, v, Wl, Wc, rpb, xa);
    mm_wm_kernel<<<576, 256, 0, stream>>>(xa, Wm, bm, xm);
    conv_kernel<<<512, 256, 0, stream>>>(xm, pp, cw, cb, bg, bb, yb);
    copy_kernel<<<4096, 256, 0, stream>>>(xm, out);
    pl_kernel  <<<64, 256, 0, stream>>>(yb, plw, plb, xm, out);
}